// ActiveParticles_74895639708064
// MI455X (gfx1250) — compile-verified
//
#include <hip/hip_runtime.h>
#include <math.h>

// ---------------------------------------------------------------------------
// ActiveParticles step for MI455X (gfx1250, wave32).
// Pairwise masked reductions (rr@pos, ra@pos, ro@ori + counts) run as real
// matmuls on the matrix pipe with V_WMMA_F32_16X16X4_F32:
//   A (16x4)  = on-the-fly mask tile (each lane owns 2 (i,j) pairs)
//   B (4x16)  = [pos_re, pos_im, 1, 0...] or [ori_re, ori_im, 0...]
//               built branchlessly from per-lane one-hot selectors (no EXEC
//               divergence in the hot loop).
//   D (16x16) = f32 accumulator, columns 0..2 hold the useful sums.
// angle_diff is branchless (atan2 angles are in [-pi,pi], so python mod(d,pi)
// for d<=-pi is exactly (d==-pi||d==-2pi) ? 0 : d+2pi).
// The collision while-loop is replaced by 3 fixed collision_step passes
// (collision_step is an exact no-op once converged).
// ---------------------------------------------------------------------------

#define N_PART   4096
#define TILES    (N_PART / 16)       // 256 row tiles
#define J_SPLITS 8
#define JCHUNK   (N_PART / J_SPLITS) // 512 j's per block
#define WAVES_PB 4                   // 4 waves (4 row-tiles) share one j stage
#define NSLOT    8                   // rr.re rr.im n_r ra.re ra.im n_a os.re os.im

typedef __attribute__((ext_vector_type(2))) float v2f;
typedef __attribute__((ext_vector_type(8))) float v8f;

__device__ __forceinline__ float angle_diff_dev(float a, float b) {
    const float PI_F  = 3.1415927410125732f;
    const float TPI_F = 6.2831854820251465f;
    float d = a - b;                 // d in [-2pi, 2pi]
    if (d <= -PI_F) {
        // python mod(d, pi) for d in [-2pi,-pi]: 0 at the endpoints, else d+2pi
        float r = d + TPI_F;
        d = (d == -PI_F || d == -TPI_F) ? 0.0f : r;
    }
    if (d >= PI_F) d -= TPI_F;
    return d;
}

// --- compute ang = atan2(ori.im, ori.re) once ------------------------------
__global__ void k_init_ang(const float* __restrict__ ori, float* __restrict__ ang) {
    int i = blockIdx.x * blockDim.x + threadIdx.x;
    if (i < N_PART) ang[i] = atan2f(ori[2 * i + 1], ori[2 * i]);
}

// --- pairwise masked matmuls via WMMA --------------------------------------
__global__ void __launch_bounds__(WAVES_PB * 32)
k_pairwise(const float* __restrict__ pos, const float* __restrict__ ori,
           const float* __restrict__ ang, float* __restrict__ partial) {
    __shared__ float4 sj[JCHUNK];  // pos.re, pos.im, ori.re, ori.im
    __shared__ float  sa[JCHUNK];  // angle

    const int blk       = blockIdx.x;
    const int tileGroup = blk & ((TILES / WAVES_PB) - 1); // 0..63
    const int jsplit    = blk >> 6;                       // 0..7
    const int j0        = jsplit * JCHUNK;

    const float2* pos2 = (const float2*)pos;
    const float2* ori2 = (const float2*)ori;

    for (int t = threadIdx.x; t < JCHUNK; t += blockDim.x) {
        int j = j0 + t;
        float2 p = pos2[j], o = ori2[j];
        sj[t] = make_float4(p.x, p.y, o.x, o.y);
        sa[t] = ang[j];
    }
    __syncthreads();

    const int wave = threadIdx.x >> 5;
    const int lane = threadIdx.x & 31;
    const int tile = tileGroup * WAVES_PB + wave;
    const int row  = lane & 15;      // A row M / B,D column N for this lane
    const int half = lane >> 4;      // K-half for A/B, M-half for D
    const int col  = row;
    const int iRow = tile * 16 + row;
    const int selfL = iRow - j0;     // local j index that equals iRow (may be OOR)

    // one-hot column selectors: B built with FMAs, never with divergent selects
    const float c0 = (col == 0) ? 1.0f : 0.0f;
    const float c1 = (col == 1) ? 1.0f : 0.0f;
    const float c2 = (col == 2) ? 1.0f : 0.0f;

    const float2 pi2 = pos2[iRow];
    const float  pix = pi2.x, piy = pi2.y;
    const float  ai  = ang[iRow];

    v8f acc_rr = {0.f,0.f,0.f,0.f,0.f,0.f,0.f,0.f};
    v8f acc_ra = acc_rr;
    v8f acc_ro = acc_rr;

    const float RR2     = 6.4e-11f;        // RR^2 = (8e-6)^2
    const float RO2     = 7.9243225e-10f;  // (RO+RC)^2 = (2.815e-5)^2
    const float ALPHA_F = 3.9269907f;      // 225 deg
    const float HPI     = 1.5707963705f;

    #pragma unroll 2
    for (int c = 0; c < JCHUNK / 4; ++c) {
        const int jb = c * 4 + half * 2;   // this lane's two K slots
        const float4 q0 = sj[jb];
        const float4 q1 = sj[jb + 1];
        const float2 aa = *(const float2*)&sa[jb];  // single ds_load_b64

        float arr[2], ara[2], aro[2], bpos[2], bori[2];
        const float4 qq[2] = {q0, q1};
        const float  aj2[2] = {aa.x, aa.y};
        #pragma unroll
        for (int v = 0; v < 2; ++v) {
            const float4 q  = qq[v];
            const float  aj = aj2[v];
            const float  dx = pix - q.x, dy = piy - q.y;
            const float  d2 = dx * dx + dy * dy;
            const float adf  = fabsf(angle_diff_dev(ai, aj));
            const bool  wv   = adf < ALPHA_F;   // within view cone
            const bool  infr = adf < HPI;       // in front
            const bool  self = (jb + v) == selfL;
            arr[v] = (d2 <= RR2 && !self && infr) ? 1.0f : 0.0f;
            ara[v] = wv ? 1.0f : 0.0f;          // RA=1e30 -> distance test always true
            aro[v] = (d2 <= RO2 && wv) ? 1.0f : 0.0f;
            bpos[v] = fmaf(q.x, c0, fmaf(q.y, c1, c2));  // branchless B build
            bori[v] = fmaf(q.z, c0, q.w * c1);
        }
        v2f A, B;
        A[0] = arr[0]; A[1] = arr[1];
        B[0] = bpos[0]; B[1] = bpos[1];
        acc_rr = __builtin_amdgcn_wmma_f32_16x16x4_f32(false, A, false, B,
                                                       (short)0, acc_rr, false, false);
        A[0] = ara[0]; A[1] = ara[1];
        acc_ra = __builtin_amdgcn_wmma_f32_16x16x4_f32(false, A, false, B,
                                                       (short)0, acc_ra, false, false);
        A[0] = aro[0]; A[1] = aro[1];
        B[0] = bori[0]; B[1] = bori[1];
        acc_ro = __builtin_amdgcn_wmma_f32_16x16x4_f32(false, A, false, B,
                                                       (short)0, acc_ro, false, false);
    }

    // D layout: lane holds column N=col; VGPR r holds row M = r + 8*half.
    float* base = partial + (size_t)(jsplit * TILES + tile) * (NSLOT * 16);
    if (col < 3) {
        #pragma unroll
        for (int r = 0; r < 8; ++r) base[(0 + col) * 16 + half * 8 + r] = acc_rr[r];
        #pragma unroll
        for (int r = 0; r < 8; ++r) base[(3 + col) * 16 + half * 8 + r] = acc_ra[r];
    }
    if (col < 2) {
        #pragma unroll
        for (int r = 0; r < 8; ++r) base[(6 + col) * 16 + half * 8 + r] = acc_ro[r];
    }
}

// --- per-particle trig + outputs -------------------------------------------
__global__ void k_finalize(const float* __restrict__ pos,  const float* __restrict__ ori,
                           const float* __restrict__ ang,  const float* __restrict__ deltas,
                           const float* __restrict__ rotn, const float* __restrict__ transn,
                           const float* __restrict__ partial,
                           float* __restrict__ out, float* __restrict__ cand) {
    int i = blockIdx.x * blockDim.x + threadIdx.x;
    if (i >= N_PART) return;
    const int tile = i >> 4, m = i & 15;

    float s[NSLOT];
    #pragma unroll
    for (int k = 0; k < NSLOT; ++k) s[k] = 0.0f;
    for (int js = 0; js < J_SPLITS; ++js) {
        const float* b = partial + (size_t)(js * TILES + tile) * (NSLOT * 16);
        #pragma unroll
        for (int k = 0; k < NSLOT; ++k) s[k] += b[k * 16 + m];
    }

    const float px = pos[2 * i], py = pos[2 * i + 1];
    const float ox = ori[2 * i], oy = ori[2 * i + 1];
    const float nr = s[2], na = s[5];
    const float snr = nr > 0.0f ? 1.0f : 0.0f;
    const float sna = na > 0.0f ? 1.0f : 0.0f;
    const float inr = 1.0f / fmaxf(nr, 1.0f);
    const float ina = 1.0f / fmaxf(na, 1.0f);

    const float Sx = s[0] * inr - px * snr, Sy = s[1] * inr - py * snr;
    const float dx = -Sx, dy = -Sy;                        // d = -S
    const float Px = s[3] * ina - px * sna, Py = s[4] * ina - py * sna;  // Ps
    const float osx = s[6], osy = s[7];                    // osum

    float sd, cd;
    sincosf(deltas[i], &sd, &cd);
    const float lx = Px * cd - Py * sd, ly = Px * sd + Py * cd;   // Ps*exp(+i d)
    const float rx = Px * cd + Py * sd, ry = Py * cd - Px * sd;   // Ps*exp(-i d)

    const float EPSF = 1e-14f;
    const float magO = fmaxf(sqrtf(osx * osx + osy * osy), EPSF);
    const float magL = fmaxf(sqrtf(lx * lx + ly * ly), EPSF);
    const float magR = fmaxf(sqrtf(rx * rx + ry * ry), EPSF);
    const float csl  = (lx * osx + ly * osy) / (magL * magO);
    const float csr  = (rx * osx + ry * osy) / (magR * magO);
    const bool  leftc = csl >= csr;                        // T >= T_RELEASE branch
    const float bx = leftc ? lx : rx, by = leftc ? ly : ry;

    const float ai  = ang[i];
    const bool  dnz = sqrtf(dx * dx + dy * dy) > 0.0f;
    const bool  bnz = sqrtf(bx * bx + by * by) > 0.0f;
    const float angSel = dnz ? atan2f(dy, dx) : (bnz ? atan2f(by, bx) : 0.0f);
    const float att    = angle_diff_dev(angSel, ai);

    // DT*GAMMA*DR = 0.014 ; sqrt(2*DR)*sqrt(DT) = 0.033466401
    const float theta = 0.014f * sinf(att) + rotn[i] * 0.033466401f;
    float sr, cr;
    sincosf(theta, &sr, &cr);
    const float nox = ox * cr - oy * sr, noy = ox * sr + oy * cr;  // ori*rotation

    // DT*VEL = 1e-7 ; sqrt(2*DTRANS)*sqrt(DT) = 7.4833095e-8
    const float tx = 1e-7f * ox + 7.4833095e-8f * transn[2 * i];
    const float ty = 1e-7f * oy + 7.4833095e-8f * transn[2 * i + 1];

    cand[2 * i]     = px + tx;       // candidate position before collision solve
    cand[2 * i + 1] = py + ty;
    out[(1 * N_PART + i) * 2 + 0] = nox; out[(1 * N_PART + i) * 2 + 1] = noy;
    out[(2 * N_PART + i) * 2 + 0] = osx; out[(2 * N_PART + i) * 2 + 1] = osy;
    out[(3 * N_PART + i) * 2 + 0] = lx;  out[(3 * N_PART + i) * 2 + 1] = ly;
    out[(4 * N_PART + i) * 2 + 0] = rx;  out[(4 * N_PART + i) * 2 + 1] = ry;
}

// --- one collision_step (identity once converged) --------------------------
#define CTILE 512
__global__ void __launch_bounds__(256)
k_collide(const float* __restrict__ pin, float* __restrict__ pout) {
    __shared__ float2 sp[CTILE];
    const int i = blockIdx.x * blockDim.x + threadIdx.x;
    const float2* pin2 = (const float2*)pin;
    const float2 p = pin2[i];
    float ax = 0.0f, ay = 0.0f;
    const float TWO_RC = 6.3e-6f;   // 2*RC
    const float OV     = 6.615e-6f; // 2.1*RC
    for (int jt = 0; jt < N_PART; jt += CTILE) {
        __syncthreads();
        for (int t = threadIdx.x; t < CTILE; t += blockDim.x)
            sp[t] = pin2[jt + t];
        __syncthreads();
        for (int jl = 0; jl < CTILE; ++jl) {
            const float2 q  = sp[jl];
            const float dcx = q.x - p.x, dcy = q.y - p.y;  // dc[i,j] = p_j - p_i
            const float da  = sqrtf(dcx * dcx + dcy * dcy);
            const float thr = ((jt + jl) == i) ? (TWO_RC - 1.0f) : TWO_RC;
            if (da <= thr) {
                const float safe = da > 0.0f ? da : 1.0f;
                const float w    = (OV - da) * 0.5f / safe;
                ax += dcx * w;
                ay += dcy * w;
            }
        }
    }
    pout[2 * i]     = p.x - ax;
    pout[2 * i + 1] = p.y - ay;
}

extern "C" void kernel_launch(void* const* d_in, const int* in_sizes, int n_in,
                              void* d_out, int out_size, void* d_ws, size_t ws_size,
                              hipStream_t stream) {
    const float* positions    = (const float*)d_in[0];
    const float* orientations = (const float*)d_in[1];
    const float* deltas       = (const float*)d_in[2];
    const float* rot_noise    = (const float*)d_in[3];
    const float* trans_noise  = (const float*)d_in[4];
    float* out = (float*)d_out;
    float* ws  = (float*)d_ws;

    // workspace layout (floats): ang[4096] | partial[8*256*128] | cand0[8192] | cand1[8192]
    float* ang     = ws;
    float* partial = ang + N_PART;
    float* cand0   = partial + (size_t)J_SPLITS * TILES * NSLOT * 16;
    float* cand1   = cand0 + 2 * N_PART;

    k_init_ang<<<N_PART / 256, 256, 0, stream>>>(orientations, ang);
    k_pairwise<<<(TILES / WAVES_PB) * J_SPLITS, WAVES_PB * 32, 0, stream>>>(
        positions, orientations, ang, partial);
    k_finalize<<<N_PART / 256, 256, 0, stream>>>(
        positions, orientations, ang, deltas, rot_noise, trans_noise,
        partial, out, cand0);
    // fixed-count collision solve; extra steps are exact no-ops once converged
    k_collide<<<N_PART / 256, 256, 0, stream>>>(cand0, cand1);
    k_collide<<<N_PART / 256, 256, 0, stream>>>(cand1, cand0);
    k_collide<<<N_PART / 256, 256, 0, stream>>>(cand0, out + 0); // out[0] = new_pos
}